// RewardCLIP_19902878449928
// MI455X (gfx1250) — compile-verified
//
#include <hip/hip_runtime.h>
#include <hip/hip_bf16.h>

typedef __attribute__((ext_vector_type(2))) float v2f;
typedef __attribute__((ext_vector_type(8))) float v8f;

#define N_ROWS          65536
#define DIMS            1024
#define ROWS_PER_BLOCK  128     // 8 waves x 16 rows/wave
#define KC              32      // k-chunk staged in LDS per iteration
#define LSTRIDE         36      // padded LDS row stride (floats): bank-conflict-free

__global__ __launch_bounds__(256) void RewardCLIP_zero_kernel(float* out) {
    if (threadIdx.x == 0 && blockIdx.x == 0) out[0] = 0.0f;
}

__global__ __launch_bounds__(256) void RewardCLIP_main_kernel(
    const float* __restrict__ inp,   // input_embs      [N, D]
    const float* __restrict__ ng,    // neg_input_embs  [N, D]
    const float* __restrict__ img,   // image_embs      [N, D]
    const int*   __restrict__ alphaP,
    float*       __restrict__ out)
{
    __shared__ float sDif[ROWS_PER_BLOCK * LSTRIDE];
    __shared__ float sImg[ROWS_PER_BLOCK * LSTRIDE];
    __shared__ float sBlockSum;

    const int t    = threadIdx.x;
    const int lane = t & 31;
    const int w    = t >> 5;                       // wave id, 0..7
    const size_t rowBase = (size_t)blockIdx.x * ROWS_PER_BLOCK;

    // WMMA f32 16x16x4 A/B per-lane layout: lane L holds row M = L%16,
    // K-offset (L>=16 ? 2 : 0), two consecutive K values per operand pair.
    const int m      = lane & 15;
    const int kofs   = (lane >> 4) << 1;           // 0 or 2
    const int ldsRow = (w * 16 + m) * LSTRIDE;

    v8f acc0 = {};                                  // 16x16 f32 accumulators
    v8f acc1 = {};

    for (int c = 0; c < DIMS / KC; ++c) {
        const int k0 = c * KC;
        __syncthreads();                            // previous compute done before restage

        // Stage: 128 rows x KC floats of dif = in - ng, and of img.
        // 1024 float4 per matrix chunk; 256 threads -> 4 each; 8 threads per
        // row give 128B contiguous segments (coalesced b128 loads).
        #pragma unroll
        for (int i = 0; i < 4; ++i) {
            const int e   = t + i * 256;            // 0..1023
            const int row = e >> 3;                 // 0..127
            const int col = (e & 7) << 2;           // 0,4,...,28
            const size_t g = (rowBase + row) * (size_t)DIMS + (size_t)(k0 + col);
            const float4 a  = *(const float4*)(inp + g);
            const float4 b  = *(const float4*)(ng  + g);
            const float4 c4 = *(const float4*)(img + g);
            float4 d4;
            d4.x = a.x - b.x; d4.y = a.y - b.y; d4.z = a.z - b.z; d4.w = a.w - b.w;
            const int l = row * LSTRIDE + col;      // byte addr multiple of 16
            *(float4*)(&sDif[l]) = d4;
            *(float4*)(&sImg[l]) = c4;
        }
        __syncthreads();

        // Compute: D += A(16x4 of dif) x B(4x16 = img^T). Same 8B/lane fetch
        // pattern feeds both operands. Two accumulators break the RAW chain.
        #pragma unroll
        for (int kk = 0; kk < KC; kk += 8) {
            v2f a0 = *(const v2f*)(&sDif[ldsRow + kk + kofs]);
            v2f b0 = *(const v2f*)(&sImg[ldsRow + kk + kofs]);
            acc0 = __builtin_amdgcn_wmma_f32_16x16x4_f32(
                false, a0, false, b0, (short)0, acc0, false, false);
            v2f a1 = *(const v2f*)(&sDif[ldsRow + kk + 4 + kofs]);
            v2f b1 = *(const v2f*)(&sImg[ldsRow + kk + 4 + kofs]);
            acc1 = __builtin_amdgcn_wmma_f32_16x16x4_f32(
                false, a1, false, b1, (short)0, acc1, false, false);
        }
    }

    // Extract diagonal of the 16x16 result tile.
    // C/D layout: VGPR r, lanes 0-15 -> (M=r, N=lane); lanes 16-31 -> (M=8+r, N=lane-16).
    // Diagonal (M==N): lane r (r<8) holds it in element r; lane 24+r holds (M=8+r) in element r.
    float delta = 0.0f;
    #pragma unroll
    for (int r = 0; r < 8; ++r) {
        const float v = acc0[r] + acc1[r];
        delta = (lane == r)      ? v : delta;
        delta = (lane == r + 24) ? v : delta;
    }
    const bool valid = (lane < 8) || (lane >= 24);

    if (t == 0) sBlockSum = 0.0f;
    __syncthreads();

    const float alphaF = (float)alphaP[0];
    const float x  = delta * alphaF;
    // log_sigmoid(x) = min(x,0) - log1p(exp(-|x|))  (numerically stable)
    float ls = valid ? (fminf(x, 0.0f) - log1pf(expf(-fabsf(x)))) : 0.0f;

    // wave32 reduction
    #pragma unroll
    for (int off = 16; off >= 1; off >>= 1)
        ls += __shfl_xor(ls, off, 32);

    if (lane == 0) atomicAdd(&sBlockSum, ls);
    __syncthreads();

    if (t == 0) atomicAdd(out, -sBlockSum * (1.0f / (float)N_ROWS));
}

extern "C" void kernel_launch(void* const* d_in, const int* in_sizes, int n_in,
                              void* d_out, int out_size, void* d_ws, size_t ws_size,
                              hipStream_t stream) {
    const float* inp   = (const float*)d_in[0];
    const float* ng    = (const float*)d_in[1];
    const float* img   = (const float*)d_in[2];
    const int*   alpha = (const int*)d_in[3];
    float* out = (float*)d_out;

    RewardCLIP_zero_kernel<<<1, 256, 0, stream>>>(out);
    RewardCLIP_main_kernel<<<N_ROWS / ROWS_PER_BLOCK, 256, 0, stream>>>(
        inp, ng, img, alpha, out);
}